// RelaxedFacenetLoss_12395275616913
// MI455X (gfx1250) — compile-verified
//
#include <hip/hip_runtime.h>
#include <math.h>

// Problem constants (match reference)
#define N_PTS      320
#define D_DIM      128
#define RHO_C      10.0f
#define MARGIN_C   0.2f
#define TAU_C      1.3f
#define SEMIHARD_T 0.01f
#define EPS_C      1e-12f

typedef float v2f __attribute__((ext_vector_type(2)));
typedef float v8f __attribute__((ext_vector_type(8)));

// Workspace layout (float offsets):
//   Xn     : 3 * 320 * 128 = 122880   (normalized rows: z=0,1 sources, z=2 embeddings)
//   sq     : 3 * 320       = 960      (||xn||^2 per row, matches reference's sq after clip)
//   dmat   : 3 * 320*320   = 307200   (pairwise distance matrices per z)
//   pair_w : 320*320       = 102400
//   num_ij : 320*320       = 102400   (pos * pair_w, deterministic per (i,j))
#define WS_XN     0
#define WS_SQ     122880
#define WS_DMAT   123840
#define WS_PAIRW  431040
#define WS_NUMIJ  533440

// ---------------------------------------------------------------------------
// Kernel 1: L2-normalize each row of the 3 matrices; one wave (32 lanes) per row.
// ---------------------------------------------------------------------------
__global__ void rfl_normalize_rows(const float* __restrict__ src,
                                   const float* __restrict__ emb,
                                   float* __restrict__ Xn,
                                   float* __restrict__ sq) {
    const int row = blockIdx.x;     // 0..319
    const int z   = blockIdx.y;     // 0,1 = sources; 2 = embeddings
    const int t   = threadIdx.x;    // 0..31 (wave32)

    const float* x = (z < 2) ? (src + (size_t)(z * N_PTS + row) * D_DIM)
                             : (emb + (size_t)row * D_DIM);

    float v[4];
    float s = 0.0f;
#pragma unroll
    for (int i = 0; i < 4; ++i) {
        v[i] = x[t + 32 * i];
        s += v[i] * v[i];
    }
    // wave32 butterfly reduction
#pragma unroll
    for (int off = 16; off > 0; off >>= 1)
        s += __shfl_xor(s, off, 32);

    const float nrm     = sqrtf(s);
    const float clipped = fmaxf(nrm, EPS_C);
    const float inv     = 1.0f / clipped;

    float* xo = Xn + (size_t)(z * N_PTS + row) * D_DIM;
#pragma unroll
    for (int i = 0; i < 4; ++i)
        xo[t + 32 * i] = v[i] * inv;

    if (t == 0)
        sq[z * N_PTS + row] = s * inv * inv;   // == sum(xn*xn)
}

// ---------------------------------------------------------------------------
// Kernel 2: Gram tile via V_WMMA_F32_16X16X4_F32, then distance transform.
// One wave per 16x16 output tile; K=128 in steps of 4 (32 WMMAs).
// A-frag and B-frag use the same per-lane addressing because B = Xn^T.
// ---------------------------------------------------------------------------
__global__ void rfl_gram_dist_wmma(const float* __restrict__ Xn,
                                   const float* __restrict__ sq,
                                   float* __restrict__ dmat) {
    const int ti = blockIdx.x;   // row tile   0..19
    const int tj = blockIdx.y;   // col tile   0..19
    const int z  = blockIdx.z;   // matrix     0..2

    const float* X = Xn + (size_t)z * N_PTS * D_DIM;

    const int lane  = threadIdx.x;   // 0..31, EXEC all ones (no divergence)
    const int m     = lane & 15;     // row within tile (A) / col within tile (B)
    const int khalf = lane >> 4;     // K sub-group select

    // Lane base pointers: element r of the float2 is K = kbase + 2*khalf + r.
    const float* arow = X + (size_t)(ti * 16 + m) * D_DIM + 2 * khalf;
    const float* brow = X + (size_t)(tj * 16 + m) * D_DIM + 2 * khalf;

    v8f acc = {};
#pragma unroll 4
    for (int kb = 0; kb < D_DIM; kb += 4) {
        v2f a = *(const v2f*)(arow + kb);
        v2f b = *(const v2f*)(brow + kb);
        // 8 args: (neg_a, A, neg_b, B, c_mod, C, reuse_a, reuse_b)
        acc = __builtin_amdgcn_wmma_f32_16x16x4_f32(
            false, a, false, b, (short)0, acc, false, false);
    }

    // d[i][j] = sqrt(max(sq_i + sq_j - 2*dot, 0) + EPS)
    const float* sqz = sq + z * N_PTS;
    float* dz = dmat + (size_t)z * N_PTS * N_PTS;

    const int   gj  = tj * 16 + m;       // C/D: N = lane & 15
    const float sqj = sqz[gj];
#pragma unroll
    for (int r = 0; r < 8; ++r) {
        const int   gi = ti * 16 + r + 8 * khalf;  // C/D: M = vgpr + 8*half
        const float d2 = sqz[gi] + sqj - 2.0f * acc[r];
        dz[(size_t)gi * N_PTS + gj] = sqrtf(fmaxf(d2, 0.0f) + EPS_C);
    }
}

// ---------------------------------------------------------------------------
// Kernel 3: average the two source distance matrices -> sigmoid pair weights.
// ---------------------------------------------------------------------------
__global__ void rfl_avg_pairw(const float* __restrict__ dmat,
                              float* __restrict__ pair_w) {
    const int idx = blockIdx.x * blockDim.x + threadIdx.x;
    if (idx >= N_PTS * N_PTS) return;
    const float avg = 0.5f * (dmat[idx] + dmat[N_PTS * N_PTS + idx]);
    const float x   = RHO_C * (TAU_C - avg);
    pair_w[idx] = 1.0f / (1.0f + expf(-x));
}

// ---------------------------------------------------------------------------
// Kernel 4: N^3 triplet reduction. One thread per (i,j), loop over k.
// d row and pair_w row (320 floats each) stay L2/L0-resident.
// ---------------------------------------------------------------------------
__global__ void rfl_triplet(const float* __restrict__ dmat,
                            const float* __restrict__ pair_w,
                            float* __restrict__ num_ij) {
    const int idx = blockIdx.x * blockDim.x + threadIdx.x;
    if (idx >= N_PTS * N_PTS) return;
    const int i = idx / N_PTS;
    const int j = idx - i * N_PTS;

    const float* drow  = dmat + (size_t)2 * N_PTS * N_PTS + (size_t)i * N_PTS; // z=2
    const float* pwrow = pair_w + (size_t)i * N_PTS;

    const float dij  = drow[j];
    const float pwij = pwrow[j];

    float maxv = -INFINITY;  // pos_semihard = max_k tl*semihard
    float minv =  INFINITY;  // norm_l       = min_k (tl - RHO*hard)
    for (int k = 0; k < N_PTS; ++k) {
        const float tl = fmaxf(MARGIN_C - (drow[k] - dij), 0.0f);
        const float tw = pwij * (1.0f - pwrow[k]);
        const float csh = (tl > 0.0f && tl <= MARGIN_C) ? tl * tw : 0.0f;
        maxv = fmaxf(maxv, csh);
        const float ch = tl - ((tl > MARGIN_C) ? RHO_C * tw : 0.0f);
        minv = fminf(minv, ch);
    }

    const float pos_hard = (minv < 0.0f) ? (minv + RHO_C) : 0.0f;
    const float pos = maxv + ((maxv <= SEMIHARD_T) ? pos_hard : 0.0f);
    num_ij[idx] = pos * pwij;
}

// ---------------------------------------------------------------------------
// Kernel 5: deterministic final reduction (single block tree reduce) + ratio.
// ---------------------------------------------------------------------------
__global__ void rfl_final_reduce(const float* __restrict__ num_ij,
                                 const float* __restrict__ pair_w,
                                 float* __restrict__ out) {
    __shared__ float snum[256];
    __shared__ float sden[256];
    const int t = threadIdx.x;
    float sn = 0.0f, sd = 0.0f;
    for (int idx = t; idx < N_PTS * N_PTS; idx += 256) {
        sn += num_ij[idx];
        sd += pair_w[idx];
    }
    snum[t] = sn;
    sden[t] = sd;
    __syncthreads();
    for (int off = 128; off > 0; off >>= 1) {
        if (t < off) {
            snum[t] += snum[t + off];
            sden[t] += sden[t + off];
        }
        __syncthreads();
    }
    if (t == 0) {
        const float den = sden[0];
        out[0] = (den > 0.0f) ? (snum[0] / fmaxf(den, EPS_C)) : 0.0f;
    }
}

// ---------------------------------------------------------------------------
extern "C" void kernel_launch(void* const* d_in, const int* in_sizes, int n_in,
                              void* d_out, int out_size, void* d_ws, size_t ws_size,
                              hipStream_t stream) {
    const float* src = (const float*)d_in[0];   // [2,320,128] f32
    const float* emb = (const float*)d_in[1];   // [320,128]   f32
    float* out = (float*)d_out;                 // scalar f32

    float* ws      = (float*)d_ws;
    float* Xn      = ws + WS_XN;
    float* sq      = ws + WS_SQ;
    float* dmat    = ws + WS_DMAT;
    float* pair_w  = ws + WS_PAIRW;
    float* num_ij  = ws + WS_NUMIJ;

    // 1) normalize rows of all three matrices (one wave per row)
    rfl_normalize_rows<<<dim3(N_PTS, 3), 32, 0, stream>>>(src, emb, Xn, sq);

    // 2) pairwise distances via fp32 WMMA Gram tiles (one wave per 16x16 tile)
    rfl_gram_dist_wmma<<<dim3(N_PTS / 16, N_PTS / 16, 3), 32, 0, stream>>>(Xn, sq, dmat);

    // 3) averaged-source sigmoid pair weights
    const int total = N_PTS * N_PTS;
    rfl_avg_pairw<<<(total + 255) / 256, 256, 0, stream>>>(dmat, pair_w);

    // 4) N^3 semihard/hard triplet reduction over k
    rfl_triplet<<<(total + 255) / 256, 256, 0, stream>>>(dmat, pair_w, num_ij);

    // 5) deterministic scalar reduction + safe division
    rfl_final_reduce<<<1, 256, 0, stream>>>(num_ij, pair_w, out);
}